// GCN_28518582845759
// MI455X (gfx1250) — compile-verified
//
#include <hip/hip_runtime.h>
#include <math.h>

// ---------------------------------------------------------------------------
// GCN 2-layer forward for gfx1250 (MI455X), fp32 WMMA (V_WMMA_F32_16X16X4_F32)
//
// Assumptions (documented, compile-only environment):
//  * A-matrix 16x4 f32 frag: lanes 0-15 hold K=0 (v0) / K=1 (v1), lanes 16-31
//    hold K=2 / K=3, M = lane&15  (ISA 7.12.2).
//  * B-matrix 4x16 f32 frag mirrors the K split: lanes 0-15 K=0/1, lanes
//    16-31 K=2/3, N = lane&15.
//  * C/D 16x16 f32: VGPR i = row i (lanes 0-15) and row i+8 (lanes 16-31),
//    N = lane&15.
//  * N_NODES (100000) is a multiple of 16 -> no M-tail.
// ---------------------------------------------------------------------------

typedef __attribute__((ext_vector_type(2))) float v2f;
typedef __attribute__((ext_vector_type(8))) float v8f;

#define WMMA_F32X4(a, b, c) \
  __builtin_amdgcn_wmma_f32_16x16x4_f32(false, (a), false, (b), (short)0, (c), false, false)

// ------------------------------ small utility kernels ----------------------

__global__ __launch_bounds__(256) void k_zero_f32(float* __restrict__ p, size_t n) {
  size_t i = (size_t)blockIdx.x * 256u + threadIdx.x;
  if (i < n) p[i] = 0.0f;
}

// degree via fp32 atomics (exact for counts < 2^24)
__global__ __launch_bounds__(256) void k_edge_deg(const long long* __restrict__ dst,
                                                  float* __restrict__ deg, size_t E) {
  size_t e = (size_t)blockIdx.x * 256u + threadIdx.x;
  if (e < E) atomicAdd(&deg[dst[e]], 1.0f);
}

// dinv = rsqrt(deg + 1)   (+1 = self loop; always > 0, matches reference where())
__global__ __launch_bounds__(256) void k_dinv(float* __restrict__ d, size_t N) {
  size_t i = (size_t)blockIdx.x * 256u + threadIdx.x;
  if (i < N) d[i] = rsqrtf(d[i] + 1.0f);
}

// ------------------------------ GEMM1: h1 = x @ W1 (512 -> 16) -------------

__global__ __launch_bounds__(256) void k_gemm1_wmma(const float* __restrict__ x,
                                                    const float* __restrict__ W1,
                                                    float* __restrict__ h1,
                                                    int nTiles) {
  __shared__ float lW[512 * 16];  // 32 KB
  for (int i = threadIdx.x; i < 512 * 16; i += 256) lW[i] = W1[i];
  __syncthreads();

  const int lane = threadIdx.x & 31;
  const int wave = threadIdx.x >> 5;
  const int tile = blockIdx.x * 8 + wave;
  if (tile >= nTiles) return;  // wave-uniform: EXEC stays all-ones for WMMA

  const int row0 = tile * 16;
  const int lc   = lane & 15;          // A row / B,C col within tile
  const int koff = (lane >> 4) << 1;   // 0 or 2: K half held by this lane
  const float* __restrict__ xr = x + (size_t)(row0 + lc) * 512 + koff;

  v8f acc = {};
  for (int k = 0; k < 512; k += 4) {
    v2f a = *(const v2f*)(xr + k);                 // x[row][k+koff .. +1]
    v2f b;
    b.x = lW[(k + koff) * 16 + lc];                // W1[k+koff][col]
    b.y = lW[(k + koff + 1) * 16 + lc];
    acc = WMMA_F32X4(a, b, acc);
  }

  const int mbase = (lane >> 4) * 8;
  for (int i = 0; i < 8; ++i)
    h1[(size_t)(row0 + mbase + i) * 16 + lc] = acc[i];
}

// agg1 = b1 + h1 * dinv^2  (self-loop term + bias)
__global__ __launch_bounds__(256) void k_init_agg1(const float* __restrict__ h1,
                                                   const float* __restrict__ dinv,
                                                   const float* __restrict__ b1,
                                                   float* __restrict__ agg1, size_t N) {
  size_t i = (size_t)blockIdx.x * 256u + threadIdx.x;  // over N*16
  if (i < N * 16) {
    size_t v = i >> 4;
    int f = (int)(i & 15);
    float di = dinv[v];
    agg1[i] = b1[f] + h1[i] * di * di;
  }
}

// edge scatter, 16 features per edge
__global__ __launch_bounds__(256) void k_scatter1(const long long* __restrict__ src,
                                                  const long long* __restrict__ dst,
                                                  const float* __restrict__ h1,
                                                  const float* __restrict__ dinv,
                                                  float* __restrict__ agg1, size_t E) {
  size_t e = (size_t)blockIdx.x * 256u + threadIdx.x;
  if (e >= E) return;
  long long s = src[e], d = dst[e];
  float w = dinv[s] * dinv[d];
  const float4* __restrict__ hs = (const float4*)(h1 + (size_t)s * 16);
  float* __restrict__ ad = agg1 + (size_t)d * 16;
#pragma unroll
  for (int j = 0; j < 4; ++j) {
    float4 hv = hs[j];
    atomicAdd(&ad[j * 4 + 0], hv.x * w);
    atomicAdd(&ad[j * 4 + 1], hv.y * w);
    atomicAdd(&ad[j * 4 + 2], hv.z * w);
    atomicAdd(&ad[j * 4 + 3], hv.w * w);
  }
}

// ------------------------------ GEMM2: h2 = relu(agg1) @ W2 (16 -> 64) -----

__global__ __launch_bounds__(256) void k_gemm2_wmma(const float* __restrict__ agg1,
                                                    const float* __restrict__ W2,
                                                    float* __restrict__ h2,
                                                    int nTiles) {
  __shared__ float lW[16 * 64];  // 4 KB
  for (int i = threadIdx.x; i < 16 * 64; i += 256) lW[i] = W2[i];
  __syncthreads();

  const int lane = threadIdx.x & 31;
  const int wave = threadIdx.x >> 5;
  const int tile = blockIdx.x * 8 + wave;
  if (tile >= nTiles) return;

  const int row0 = tile * 16;
  const int lc   = lane & 15;
  const int koff = (lane >> 4) << 1;
  const float* __restrict__ ar = agg1 + (size_t)(row0 + lc) * 16 + koff;

  v8f acc0 = {}, acc1 = {}, acc2 = {}, acc3 = {};
  for (int kk = 0; kk < 16; kk += 4) {
    v2f a = *(const v2f*)(ar + kk);
    a.x = fmaxf(a.x, 0.0f);   // fused ReLU on layer-2 input
    a.y = fmaxf(a.y, 0.0f);
    const int kb = (kk + koff) * 64 + lc;
    v2f b;
    b.x = lW[kb +  0]; b.y = lW[kb + 64 +  0]; acc0 = WMMA_F32X4(a, b, acc0);
    b.x = lW[kb + 16]; b.y = lW[kb + 64 + 16]; acc1 = WMMA_F32X4(a, b, acc1);
    b.x = lW[kb + 32]; b.y = lW[kb + 64 + 32]; acc2 = WMMA_F32X4(a, b, acc2);
    b.x = lW[kb + 48]; b.y = lW[kb + 64 + 48]; acc3 = WMMA_F32X4(a, b, acc3);
  }

  const int mbase = (lane >> 4) * 8;
  for (int i = 0; i < 8; ++i) {
    size_t r = (size_t)(row0 + mbase + i) * 64;
    h2[r +  0 + lc] = acc0[i];
    h2[r + 16 + lc] = acc1[i];
    h2[r + 32 + lc] = acc2[i];
    h2[r + 48 + lc] = acc3[i];
  }
}

// agg2 = b2 + h2 * dinv^2
__global__ __launch_bounds__(256) void k_init_agg2(const float* __restrict__ h2,
                                                   const float* __restrict__ dinv,
                                                   const float* __restrict__ b2,
                                                   float* __restrict__ agg2, size_t N) {
  size_t i = (size_t)blockIdx.x * 256u + threadIdx.x;  // over N*64
  if (i < N * 64) {
    size_t v = i >> 6;
    int f = (int)(i & 63);
    float di = dinv[v];
    agg2[i] = b2[f] + h2[i] * di * di;
  }
}

// edge scatter, 64 features per edge, 4 threads per edge (16 feats each)
__global__ __launch_bounds__(256) void k_scatter2(const long long* __restrict__ src,
                                                  const long long* __restrict__ dst,
                                                  const float* __restrict__ h2,
                                                  const float* __restrict__ dinv,
                                                  float* __restrict__ agg2, size_t E) {
  size_t t = (size_t)blockIdx.x * 256u + threadIdx.x;
  size_t e = t >> 2;
  if (e >= E) return;
  int part = (int)(t & 3);
  long long s = src[e], d = dst[e];
  float w = dinv[s] * dinv[d];
  const float4* __restrict__ hs = (const float4*)(h2 + (size_t)s * 64 + part * 16);
  float* __restrict__ ad = agg2 + (size_t)d * 64 + part * 16;
#pragma unroll
  for (int j = 0; j < 4; ++j) {
    float4 hv = hs[j];
    atomicAdd(&ad[j * 4 + 0], hv.x * w);
    atomicAdd(&ad[j * 4 + 1], hv.y * w);
    atomicAdd(&ad[j * 4 + 2], hv.z * w);
    atomicAdd(&ad[j * 4 + 3], hv.w * w);
  }
}

// centers[g] = first node whose (sorted) batch id == g
__global__ __launch_bounds__(256) void k_centers(const long long* __restrict__ batch,
                                                 int* __restrict__ centers, size_t N) {
  size_t i = (size_t)blockIdx.x * 256u + threadIdx.x;
  if (i < N) {
    long long b = batch[i];
    if (i == 0 || batch[i - 1] != b) centers[b] = (int)i;
  }
}

// log_softmax over 64 features; one wave32 per graph, 2 feats/lane
__global__ __launch_bounds__(256) void k_logsoftmax(const float* __restrict__ agg2,
                                                    const int* __restrict__ centers,
                                                    float* __restrict__ out, int G) {
  int lane = threadIdx.x & 31;
  int g = (blockIdx.x * 8) + (threadIdx.x >> 5);
  if (g >= G) return;
  size_t c = (size_t)centers[g] * 64;
  float v0 = agg2[c + lane];
  float v1 = agg2[c + lane + 32];
  float m = fmaxf(v0, v1);
#pragma unroll
  for (int off = 16; off >= 1; off >>= 1) m = fmaxf(m, __shfl_xor(m, off, 32));
  float s = __expf(v0 - m) + __expf(v1 - m);
#pragma unroll
  for (int off = 16; off >= 1; off >>= 1) s += __shfl_xor(s, off, 32);
  float ls = m + __logf(s);
  out[(size_t)g * 64 + lane]      = v0 - ls;
  out[(size_t)g * 64 + lane + 32] = v1 - ls;
}

// ------------------------------ launcher -----------------------------------

extern "C" void kernel_launch(void* const* d_in, const int* in_sizes, int n_in,
                              void* d_out, int out_size, void* d_ws, size_t ws_size,
                              hipStream_t stream) {
  const float*     x   = (const float*)d_in[0];
  const long long* ei  = (const long long*)d_in[1];  // [2, E] int64
  const long long* bat = (const long long*)d_in[2];  // [N]    int64
  const float*     W1  = (const float*)d_in[4];      // [512,16]
  const float*     b1  = (const float*)d_in[5];      // [16]
  const float*     W2  = (const float*)d_in[6];      // [16,64]
  const float*     b2  = (const float*)d_in[7];      // [64]
  float*           out = (float*)d_out;

  const size_t N = (size_t)in_sizes[0] / 512;        // 100000
  const size_t E = (size_t)in_sizes[1] / 2;          // 3200000
  const int    G = out_size / 64;                    // 1024

  const long long* src = ei;
  const long long* dst = ei + E;

  // workspace carve-out (all float unless noted), 64-element aligned
  size_t off = 0;
  auto carve = [&](size_t nElems) { size_t p = off; off += (nElems + 63) & ~(size_t)63; return p; };
  float* ws   = (float*)d_ws;
  float* dinv = ws + carve(N);           // deg, then dinv in place
  float* h1   = ws + carve(N * 16);
  float* agg1 = ws + carve(N * 16);
  float* h2   = ws + carve(N * 64);
  float* agg2 = ws + carve(N * 64);
  int*   ctr  = (int*)(ws + carve((size_t)G));
  (void)ws_size; (void)n_in; (void)bat;

  const int nTiles = (int)((N + 15) / 16);           // 6250 (N % 16 == 0)
  auto blocks = [](size_t work) { return (unsigned)((work + 255) / 256); };

  // 1) degree + dinv (self-loop folded in as +1)
  k_zero_f32<<<blocks(N), 256, 0, stream>>>(dinv, N);
  k_edge_deg<<<blocks(E), 256, 0, stream>>>(dst, dinv, E);
  k_dinv<<<blocks(N), 256, 0, stream>>>(dinv, N);

  // 2) layer 1
  k_gemm1_wmma<<<(nTiles + 7) / 8, 256, 0, stream>>>(x, W1, h1, nTiles);
  k_init_agg1<<<blocks(N * 16), 256, 0, stream>>>(h1, dinv, b1, agg1, N);
  k_scatter1<<<blocks(E), 256, 0, stream>>>(src, dst, h1, dinv, agg1, E);

  // 3) layer 2 (ReLU fused into GEMM2 A-fragment load)
  k_gemm2_wmma<<<(nTiles + 7) / 8, 256, 0, stream>>>(agg1, W2, h2, nTiles);
  k_init_agg2<<<blocks(N * 64), 256, 0, stream>>>(h2, dinv, b2, agg2, N);
  k_scatter2<<<blocks(E * 4), 256, 0, stream>>>(src, dst, h2, dinv, agg2, E);

  // 4) readout
  k_centers<<<blocks(N), 256, 0, stream>>>(bat, ctr, N);
  k_logsoftmax<<<(unsigned)((G + 7) / 8), 256, 0, stream>>>(agg2, ctr, out, G);
}